// tree_crf_70248485093749
// MI455X (gfx1250) — compile-verified
//
#include <hip/hip_runtime.h>

// Problem constants from the reference.
#define BATCH 32
#define MDIM  64
#define RDIM  8
#define NPAIR (MDIM * MDIM)          // 4096 (i,j) pairs
#define NELEM (NPAIR * RDIM)         // 32768 floats per batch slice
#define BLK   1024                   // 32 wave32 waves per workgroup

typedef unsigned int v4u __attribute__((ext_vector_type(4)));
typedef int          v8i __attribute__((ext_vector_type(8)));
typedef int          v4i __attribute__((ext_vector_type(4)));

#define NEG_INF (-__builtin_inff())

// ds_swizzle group-of-32 xor-butterfly: offset = (xor<<10)|(and=0x1f)
#define RED_STEP(IMM) do {                                                        \
    float ov = __uint_as_float(__builtin_amdgcn_ds_swizzle(__float_as_uint(bv), (IMM))); \
    int   oi = __builtin_amdgcn_ds_swizzle(bi, (IMM));                            \
    if (ov > bv || (ov == bv && oi < bi)) { bv = ov; bi = oi; }                   \
  } while (0)

__global__ __launch_bounds__(BLK, 1)
void mst_batch_kernel(const float* __restrict__ le_g,
                      const int* __restrict__ tree_g,
                      float* __restrict__ ws) {
  __shared__ __align__(16) float s_le[NELEM];           // 128 KB TDM destination
  __shared__ unsigned long long  s_parts[MDIM];         // partition bitmasks
  __shared__ float s_redv[BLK / 32];
  __shared__ int   s_redi[BLK / 32];
  __shared__ float s_sc[MDIM];
  __shared__ unsigned long long  s_mf, s_mt;            // broadcast merge masks

  // Per-pair max overlays the (dead after setup) front of the energy slice.
  float* const s_pm = s_le;                             // 4096 floats

  const int tid  = threadIdx.x;
  const int lane = tid & 31;
  const int wid  = tid >> 5;
  const int b    = blockIdx.x;

  // ---- Async TDM fill of LDS: 1-D tile of 32768 f32 elements -------------
  if (tid == 0) {
    unsigned long long gaddr =
        (unsigned long long)(const void*)(le_g + (size_t)b * NELEM);
    unsigned lds_off = (unsigned)(unsigned long long)(void*)&s_le[0];

    v4u g0;
    g0.x = 1u;                                   // count=1 (valid), user mode
    g0.y = lds_off;                              // lds_addr (bytes)
    g0.z = (unsigned)(gaddr & 0xFFFFFFFFull);    // global_addr[31:0]
    g0.w = (unsigned)((gaddr >> 32) & 0x01FFFFFFull) | (2u << 30); // addr[56:32], type=2

    v8i g1;
    g1[0] = 0x20000;            // workgroup_mask=0, data_size=2 (4 bytes)
    g1[1] = (int)0x80000000u;   // tensor_dim0[15:0]=0x8000 in bits[31:16]
    g1[2] = 0x10000;            // tensor_dim0[31:16]=0, tensor_dim1=1
    g1[3] = (int)0x80000000u;   // tile_dim0 = 0x8000 (32768 elements)
    g1[4] = 1;                  // tile_dim1 = 1, tile_dim2 = 0
    g1[5] = NELEM;              // tensor_dim0_stride[31:0]
    g1[6] = 0;
    g1[7] = 0;

    v4i g2 = {0, 0, 0, 0};
    v4i g3 = {0, 0, 0, 0};
#if defined(__clang_major__) && (__clang_major__ >= 23)
    v8i g4 = {0, 0, 0, 0, 0, 0, 0, 0};
    __builtin_amdgcn_tensor_load_to_lds(g0, g1, g2, g3, g4, 0);
#else
    __builtin_amdgcn_tensor_load_to_lds(g0, g1, g2, g3, 0);
#endif
  }
  __builtin_amdgcn_s_wait_tensorcnt(0);
  __syncthreads();

  // ---- Setup: reduce R into per-pair max + label-score gather ------------
  // All reads of s_le happen here, into registers, before s_pm overlays it.
  float pm[4];
#pragma unroll
  for (int c = 0; c < 4; ++c) {
    int pr = c * BLK + tid;                      // pair index p*64+q
    const float* src = &s_le[pr << 3];
    float4 a = *reinterpret_cast<const float4*>(src);
    float4 d = *reinterpret_cast<const float4*>(src + 4);
    pm[c] = fmaxf(fmaxf(fmaxf(a.x, a.y), fmaxf(a.z, a.w)),
                  fmaxf(fmaxf(d.x, d.y), fmaxf(d.z, d.w)));
  }
  float scv = 0.0f;
  if (tid < MDIM) {
    const int* t3 = tree_g + ((size_t)b * MDIM + tid) * 3;
    int i = t3[0], j = t3[1], k = t3[2];
    scv = s_le[(i << 9) + (j << 3) + k];
    if ((i | j | k) == 0) scv = 0.0f;            // le.at[0,0,0].set(0)
  }
  __syncthreads();                               // all s_le reads done
#pragma unroll
  for (int c = 0; c < 4; ++c) s_pm[c * BLK + tid] = pm[c];
  if (tid < MDIM) {
    s_sc[tid]    = scv;
    s_parts[tid] = 1ull << tid;
  }
  __syncthreads();

  float score = 0.0f;
  float mst_e = 0.0f;
  if (tid == 0) {
    for (int t = 0; t < MDIM; ++t) score += s_sc[t];
  }

  // ---- 64 greedy MST steps over the 4096-entry pair-max array ------------
  for (int it = 0; it < MDIM; ++it) {
    // One float4 per thread; ascending order keeps the first (lowest flat
    // index) maximum, matching jnp.argmax tie-breaking.
    const int e = tid << 2;
    float4 v = *reinterpret_cast<const float4*>(&s_pm[e]);
    float bv = v.x; int bi = e;
    if (v.y > bv) { bv = v.y; bi = e + 1; }
    if (v.z > bv) { bv = v.z; bi = e + 2; }
    if (v.w > bv) { bv = v.w; bi = e + 3; }
    // Wave32 xor-butterfly reduction (value+index) via ds_swizzle.
    RED_STEP(0x401F); RED_STEP(0x201F); RED_STEP(0x101F);
    RED_STEP(0x081F); RED_STEP(0x041F);
    if (lane == 0) { s_redv[wid] = bv; s_redi[wid] = bi; }
    __syncthreads();

    if (tid == 0) {
      float Bv = s_redv[0]; int Bi = s_redi[0];
      for (int w = 1; w < BLK / 32; ++w) {
        float ov = s_redv[w]; int oi = s_redi[w];
        if (ov > Bv || (ov == Bv && oi < Bi)) { Bv = ov; Bi = oi; }
      }
      if (Bv != NEG_INF) mst_e += Bv;            // upd = !isneginf(max_e)
      s_mf = s_parts[Bi >> 6];                   // broadcast merge masks
      s_mt = s_parts[Bi & 63];
    }
    __syncthreads();

    const unsigned long long mf = s_mf;
    const unsigned long long mt = s_mt;
    // Block every pair (both directions) between the two merged partitions.
#pragma unroll
    for (int c = 0; c < 4; ++c) {
      int pe = c * BLK + tid;
      int p = pe >> 6, q = pe & 63;
      bool blk = (((mf >> p) & 1ull) && ((mt >> q) & 1ull)) ||
                 (((mf >> q) & 1ull) && ((mt >> p) & 1ull));
      if (blk) s_pm[pe] = NEG_INF;
    }
    // Merge: every node in the union gets the union mask. (Only thread 0
    // reads s_parts again, after the next iteration's first barrier.)
    if (tid < MDIM) {
      unsigned long long mg = mf | mt;
      if ((mg >> tid) & 1ull) s_parts[tid] = mg;
    }
    __syncthreads();                             // s_pm stable for next scan
  }

  if (tid == 0) ws[b] = mst_e - score;
}

__global__ void finalize_kernel(const float* __restrict__ ws,
                                float* __restrict__ out) {
  if (threadIdx.x == 0) {
    float s = 0.0f;
    for (int i = 0; i < BATCH; ++i) s += ws[i];
    out[0] = s * (1.0f / (float)BATCH);
  }
}

extern "C" void kernel_launch(void* const* d_in, const int* in_sizes, int n_in,
                              void* d_out, int out_size, void* d_ws, size_t ws_size,
                              hipStream_t stream) {
  const float* le   = (const float*)d_in[0];   // [32,64,64,8] f32
  const int*   tree = (const int*)d_in[1];     // [32,64,3] i32
  float* ws  = (float*)d_ws;                   // 32 floats scratch
  float* out = (float*)d_out;                  // 1 float

  mst_batch_kernel<<<BATCH, BLK, 0, stream>>>(le, tree, ws);
  finalize_kernel<<<1, 32, 0, stream>>>(ws, out);
}